// ShortConv_3934190044142
// MI455X (gfx1250) — compile-verified
//
#include <hip/hip_runtime.h>

// ShortConv fused: RMSNorm(D) -> depthwise causal conv1d(K=4) -> SiLU
// B=4, L=4096, D=2048, K=4, fp32. Memory-bound: ~268MB min traffic.
// Single pass over x via 152KB LDS tiles per workgroup (2 WGs / 320KB WGP).
// Staging: TDM tensor_load_to_lds (one DMA per tile) with async-load fallback.

#define B_      4
#define L_      4096
#define D_      2048
#define K_      4
#define EPS_    1e-5f
#define TILE_L  16
#define HALO    (K_ - 1)            // 3
#define ROWS    (TILE_L + HALO)     // 19 rows * 8KB = 152KB LDS
#define THREADS 256                 // 8 wave32

#define AS1 __attribute__((address_space(1)))
#define AS3 __attribute__((address_space(3)))

typedef int          v4i_t __attribute__((ext_vector_type(4)));
typedef unsigned int v4u_t __attribute__((ext_vector_type(4)));
typedef int          v8i_t __attribute__((ext_vector_type(8)));

#if defined(__has_builtin)
#if __has_builtin(__builtin_amdgcn_tensor_load_to_lds)
#define USE_TDM 1
#else
#define USE_TDM 0
#endif
#if __has_builtin(__builtin_amdgcn_global_load_async_to_lds_b128)
#define USE_ASYNC_LDS 1
#else
#define USE_ASYNC_LDS 0
#endif
#if __has_builtin(__builtin_amdgcn_s_wait_asynccnt)
#define HAVE_WAIT_ASYNC_BUILTIN 1
#else
#define HAVE_WAIT_ASYNC_BUILTIN 0
#endif
#if __has_builtin(__builtin_amdgcn_s_wait_tensorcnt)
#define HAVE_WAIT_TENSOR_BUILTIN 1
#else
#define HAVE_WAIT_TENSOR_BUILTIN 0
#endif
#else
#define USE_TDM 0
#define USE_ASYNC_LDS 0
#define HAVE_WAIT_ASYNC_BUILTIN 0
#define HAVE_WAIT_TENSOR_BUILTIN 0
#endif

__device__ __forceinline__ float silu(float y) {
    return y / (1.0f + expf(-y));
}

__global__ __launch_bounds__(THREADS) void shortconv_fused_kernel(
    const float* __restrict__ x,        // (B, L, D)
    const float* __restrict__ norm_w,   // (D,)
    const float* __restrict__ conv_w,   // (K, 1, D) -> conv_w[k*D + d]
    float* __restrict__ out)            // (B, L, D)
{
    __shared__ float sx[ROWS * D_];     // raw x tile (rows l0-3 .. l0+15)
    __shared__ float sinv[ROWS];        // per-row 1/rms

    const int tid  = threadIdx.x;
    const int b    = blockIdx.y;
    const int l0   = blockIdx.x * TILE_L;
    const float* xb = x + (size_t)b * L_ * D_;

    const int gl0   = l0 - HALO;                   // first needed row (may be <0)
    const int skip  = (gl0 < 0) ? -gl0 : 0;        // causal zero-pad rows (3 or 0)
    const int nrows = ROWS - skip;                 // rows actually loaded

#if USE_TDM
    // ---- Issue one TDM DMA for the whole (nrows x D) fp32 tile FIRST, so the
    // fill runs while we zero the causal pad rows below.
    if (tid < 32) {                                // wave 0 issues; others just barrier
        const unsigned int lds_base =
            (unsigned int)(size_t)(AS3 float*)sx + (unsigned int)(skip * D_ * 4);
        const unsigned long long gaddr =
            (unsigned long long)(size_t)(xb + (size_t)(gl0 + skip) * D_);

        v4u_t g0;
        g0.x = 1u;                                            // count=1 valid descriptor
        g0.y = lds_base;                                      // lds_addr [63:32]
        g0.z = (unsigned int)gaddr;                           // global_addr lo
        g0.w = ((unsigned int)(gaddr >> 32) & 0x01FFFFFFu)    // global_addr [56:32]
               | (2u << 30);                                  // type=2 ("image")

        v8i_t g1;
        g1[0] = 0x00020000;                                   // data_size=2 (4 bytes)
        g1[1] = (int)(((unsigned)D_ & 0xFFFFu) << 16);        // tensor_dim0 lo16 @ [63:48]
        g1[2] = (int)((((unsigned)D_ >> 16) & 0xFFFFu)        // tensor_dim0 hi16
               | ((unsigned)nrows << 16));                    // tensor_dim1 lo16
        g1[3] = (int)((((unsigned)nrows >> 16) & 0xFFFFu)     // tensor_dim1 hi16 (=0)
               | (((unsigned)D_ & 0xFFFFu) << 16));           // tile_dim0 = D
        g1[4] = nrows;                                        // tile_dim1; tile_dim2=0
        g1[5] = D_;                                           // tensor_dim0_stride lo32
        g1[6] = 0;                                            // stride hi16 | dim1_stride lo16
        g1[7] = 0;                                            // dim1_stride hi32

        v4i_t g2 = {0, 0, 0, 0};                              // dims 2/3 unused
        v4i_t g3 = {0, 0, 0, 0};                              // dims 3/4 unused
        v8i_t g4 = {0, 0, 0, 0, 0, 0, 0, 0};                  // extra group (unused, 2-D tile)

        __builtin_amdgcn_tensor_load_to_lds(g0, g1, g2, g3, g4, 0);
    }
#endif

    // ---- Zero-fill causal pad rows (first tile of each batch only); overlaps DMA.
    if (skip > 0) {
        for (int c = tid; c < skip * (D_ / 4); c += THREADS)
            ((float4*)sx)[c] = make_float4(0.f, 0.f, 0.f, 0.f);
    }

#if USE_TDM
    if (tid < 32) {
#if HAVE_WAIT_TENSOR_BUILTIN
        __builtin_amdgcn_s_wait_tensorcnt(0);
#else
        asm volatile("s_wait_tensorcnt 0x0" ::: "memory");
#endif
    }
#else
    // ---- Fallback: per-lane async (or plain) staging of the loaded rows.
    const int CH0 = skip * (D_ / 4);
    const int CHUNKS4 = ROWS * (D_ / 4);
    for (int c = CH0 + tid; c < CHUNKS4; c += THREADS) {
        const int r      = c >> 9;                 // c / 512
        const int inrow4 = c & 511;                // c % 512
        float* lp = &sx[r * D_ + inrow4 * 4];
        const float* gp = xb + (size_t)(gl0 + r) * D_ + inrow4 * 4;
#if USE_ASYNC_LDS
        __builtin_amdgcn_global_load_async_to_lds_b128(
            (AS1 v4i_t*)gp, (AS3 v4i_t*)lp, 0, 0);
#else
        *(float4*)lp = *(const float4*)gp;
#endif
    }
#if USE_ASYNC_LDS
#if HAVE_WAIT_ASYNC_BUILTIN
    __builtin_amdgcn_s_wait_asynccnt(0);
#else
    asm volatile("s_wait_asynccnt 0" ::: "memory");
#endif
#endif
#endif // USE_TDM
    __syncthreads();

    // ---- Per-row RMS: one wave32 per row, rows round-robin over 8 waves.
    const int wave = tid >> 5;
    const int lane = tid & 31;
    for (int r = wave; r < ROWS; r += (THREADS / 32)) {
        const float4* row4 = (const float4*)&sx[r * D_];
        float s = 0.f;
        #pragma unroll
        for (int t = 0; t < (D_ / 4) / 32; ++t) {     // 16 float4 per lane
            float4 v = row4[lane + 32 * t];
            s += v.x * v.x + v.y * v.y + v.z * v.z + v.w * v.w;
        }
        #pragma unroll
        for (int off = 16; off > 0; off >>= 1)
            s += __shfl_xor(s, off, 32);
        if (lane == 0)
            sinv[r] = 1.0f / sqrtf(s * (1.0f / (float)D_) + EPS_);
    }
    __syncthreads();

    // ---- Conv + SiLU: each thread owns D_/(4*THREADS)=2 float4 channel groups,
    // slides a 4-deep normalized window down the tile (each LDS row read once).
    #pragma unroll
    for (int g = 0; g < D_ / (4 * THREADS); ++g) {
        const int d4 = (tid + g * THREADS) * 4;
        const float4 w0 = *(const float4*)&conv_w[0 * D_ + d4];
        const float4 w1 = *(const float4*)&conv_w[1 * D_ + d4];
        const float4 w2 = *(const float4*)&conv_w[2 * D_ + d4];
        const float4 w3 = *(const float4*)&conv_w[3 * D_ + d4];
        const float4 nw = *(const float4*)&norm_w[d4];

        // window: a = row l (oldest, tap w0), b1 = l+1, c1 = l+2, d1 = l+3
        float4 a, b1, c1, d1;
        {
            float4 v; float iv;
            v = *(const float4*)&sx[0 * D_ + d4]; iv = sinv[0];
            a  = make_float4(v.x * iv, v.y * iv, v.z * iv, v.w * iv);
            v = *(const float4*)&sx[1 * D_ + d4]; iv = sinv[1];
            b1 = make_float4(v.x * iv, v.y * iv, v.z * iv, v.w * iv);
            v = *(const float4*)&sx[2 * D_ + d4]; iv = sinv[2];
            c1 = make_float4(v.x * iv, v.y * iv, v.z * iv, v.w * iv);
        }
        float* outb = out + ((size_t)b * L_ + l0) * D_ + d4;
        #pragma unroll
        for (int l = 0; l < TILE_L; ++l) {
            const float4 v  = *(const float4*)&sx[(l + HALO) * D_ + d4];
            const float iv  = sinv[l + HALO];
            d1 = make_float4(v.x * iv, v.y * iv, v.z * iv, v.w * iv);

            float4 y;
            y.x = (w0.x * a.x + w1.x * b1.x + w2.x * c1.x + w3.x * d1.x) * nw.x;
            y.y = (w0.y * a.y + w1.y * b1.y + w2.y * c1.y + w3.y * d1.y) * nw.y;
            y.z = (w0.z * a.z + w1.z * b1.z + w2.z * c1.z + w3.z * d1.z) * nw.z;
            y.w = (w0.w * a.w + w1.w * b1.w + w2.w * c1.w + w3.w * d1.w) * nw.w;

            float4 o;
            o.x = silu(y.x); o.y = silu(y.y); o.z = silu(y.z); o.w = silu(y.w);
            *(float4*)(outb + (size_t)l * D_) = o;

            a = b1; b1 = c1; c1 = d1;
        }
    }
}

extern "C" void kernel_launch(void* const* d_in, const int* in_sizes, int n_in,
                              void* d_out, int out_size, void* d_ws, size_t ws_size,
                              hipStream_t stream) {
    const float* x      = (const float*)d_in[0];  // (B,L,D)
    const float* norm_w = (const float*)d_in[1];  // (D,)
    const float* conv_w = (const float*)d_in[2];  // (K,1,D)
    float* out = (float*)d_out;

    dim3 grid(L_ / TILE_L, B_);   // 256 x 4 = 1024 workgroups
    shortconv_fused_kernel<<<grid, THREADS, 0, stream>>>(x, norm_w, conv_w, out);
}